// PointPillarScatter_InterSweep_51582557225477
// MI455X (gfx1250) — compile-verified
//
#include <hip/hip_runtime.h>
#include <stdint.h>

// Problem constants (from reference)
constexpr int kNX = 432;
constexpr int kNY = 496;
constexpr int kB  = 4;
constexpr int kC  = 64;
constexpr int kPper = 16000;
constexpr int kNPil = kB * kPper;              // 64000 pillar rows per bin
constexpr int kMapElems = kB * kNY * kNX;      // 857088
constexpr int kCTile = 16;                     // channels per block
constexpr int kCTiles = kC / kCTile;           // 4
constexpr size_t kBinOut = (size_t)kB * kC * kNY * kNX;

typedef float v4f __attribute__((ext_vector_type(4)));
typedef int   v4i __attribute__((ext_vector_type(4)));

// Exact vector type the async-LDS builtin expects (per hipcc diagnostic).
typedef int vsi4 __attribute__((vector_size(16)));
typedef __attribute__((address_space(1))) vsi4* gvsi4_p;   // global (device) ptr
typedef __attribute__((address_space(3))) vsi4* lvsi4_p;   // LDS ptr

#if __has_builtin(__builtin_amdgcn_global_load_async_to_lds_b128)
#define HAVE_ASYNC_LDS 1
#else
#define HAVE_ASYNC_LDS 0
#endif

// ---- 1) map[*] = -1 -------------------------------------------------------
__global__ __launch_bounds__(256) void fill_map_kernel(int* __restrict__ map) {
    int i = blockIdx.x * 256 + threadIdx.x;          // one v4i per thread
    if (i < kMapElems / 4) {
        v4i m = { -1, -1, -1, -1 };
        ((v4i*)map)[i] = m;
    }
}

// ---- 2) map[flat(b,z,y,x)] = pillar row index -----------------------------
__global__ __launch_bounds__(256) void build_map_kernel(const int* __restrict__ coords,
                                                        int* __restrict__ map) {
    int p = blockIdx.x * 256 + threadIdx.x;
    if (p < kNPil) {
        v4i c = ((const v4i*)coords)[p];             // [b, z, y, x]
        int flat = c.x * (kNY * kNX) + c.y + c.z * kNX + c.w;
        map[flat] = p;                                // cells unique per construction
    }
}

// ---- 3) fused zero+gather: out[b,c,y,x] = (map>=0) ? pillars[p,c] : 0 -----
__global__ __launch_bounds__(128) void gather_kernel(const float* __restrict__ pillars,
                                                     const int* __restrict__ map,
                                                     float* __restrict__ out) {
    __shared__ __align__(16) int row[kNX];

    const int tid = threadIdx.x;
    const int blk = blockIdx.x;                       // (b*kNY + y)*kCTiles + ct
    const int ct  = blk % kCTiles;
    const int by  = blk / kCTiles;
    const int y   = by % kNY;
    const int b   = by / kNY;

    // Stage one map row (432 ints) into LDS: 108 lanes x b128.
    const int* grow = map + (size_t)(b * kNY + y) * kNX;
    if (tid < kNX / 4) {
#if HAVE_ASYNC_LDS
        __builtin_amdgcn_global_load_async_to_lds_b128(
            (gvsi4_p)(void*)(grow + 4 * tid),
            (lvsi4_p)(void*)(&row[4 * tid]),
            0, 0);
#else
        ((v4i*)row)[tid] = ((const v4i*)grow)[tid];
#endif
    }
#if HAVE_ASYNC_LDS
#if __has_builtin(__builtin_amdgcn_s_wait_asynccnt)
    __builtin_amdgcn_s_wait_asynccnt(0);
#else
    asm volatile("s_wait_asynccnt 0" ::: "memory");
#endif
#endif
    __syncthreads();

    if (tid < kNX / 4) {
        v4i pidx = ((const v4i*)row)[tid];
        const int c0 = ct * kCTile;
        float* obase = out + (((size_t)(b * kC + c0) * kNY + y) * kNX) + 4 * tid;
#pragma unroll
        for (int cc = 0; cc < kCTile; ++cc) {
            const int c = c0 + cc;
            v4f v;
            v.x = (pidx.x >= 0) ? pillars[(size_t)pidx.x * kC + c] : 0.0f;
            v.y = (pidx.y >= 0) ? pillars[(size_t)pidx.y * kC + c] : 0.0f;
            v.z = (pidx.z >= 0) ? pillars[(size_t)pidx.z * kC + c] : 0.0f;
            v.w = (pidx.w >= 0) ? pillars[(size_t)pidx.w * kC + c] : 0.0f;
            // Streaming 658 MB output: non-temporal so L2 keeps pillars/map hot.
            __builtin_nontemporal_store(v, (v4f*)(obase + (size_t)cc * kNY * kNX));
        }
    }
}

extern "C" void kernel_launch(void* const* d_in, const int* in_sizes, int n_in,
                              void* d_out, int out_size, void* d_ws, size_t ws_size,
                              hipStream_t stream) {
    (void)in_sizes; (void)n_in; (void)out_size; (void)ws_size;
    int*   map = (int*)d_ws;                    // 3.43 MB scratch, reused per bin
    float* out = (float*)d_out;

    const int fillGrid   = (kMapElems / 4 + 255) / 256;   // 837
    const int buildGrid  = (kNPil + 255) / 256;           // 250
    const int gatherGrid = kB * kNY * kCTiles;            // 7936

    for (int bin = 0; bin < 3; ++bin) {
        const float* pil    = (const float*)d_in[2 * bin + 0];
        const int*   coords = (const int*)  d_in[2 * bin + 1];
        float*       obin   = out + (size_t)bin * kBinOut;

        fill_map_kernel <<<fillGrid,   256, 0, stream>>>(map);
        build_map_kernel<<<buildGrid,  256, 0, stream>>>(coords, map);
        gather_kernel   <<<gatherGrid, 128, 0, stream>>>(pil, map, obin);
    }
}